// SAGE_79568564126324
// MI455X (gfx1250) — compile-verified
//
#include <hip/hip_runtime.h>
#include <hip/hip_bf16.h>

// GraphSAGE 2-layer on MI455X (gfx1250).
// Scatter/atomics-bound problem; dense 4->16 transform mapped onto
// V_WMMA_F32_16X16X4_F32 (exact shape match, full f32 precision).

typedef __attribute__((ext_vector_type(2))) float v2f;
typedef __attribute__((ext_vector_type(8))) float v8f;

#define NNODES 500000
#define NEDGES 5000000
#define NTILES (NNODES / 16)   // 31250, exact

// ---------------- zero workspace ----------------
__global__ void k_zero(float* __restrict__ p, int n) {
    int i = blockIdx.x * blockDim.x + threadIdx.x;
    if (i < n) p[i] = 0.0f;
}

// ---------------- degree count ----------------
__global__ void k_degree(const int* __restrict__ dst, float* __restrict__ deg) {
    int e = blockIdx.x * blockDim.x + threadIdx.x;
    if (e < NEDGES) atomicAdd(&deg[dst[e]], 1.0f);
}

// ---------------- deg -> deg_inv (in place) ----------------
__global__ void k_deg_inv(float* __restrict__ deg) {
    int i = blockIdx.x * blockDim.x + threadIdx.x;
    if (i < NNODES) {
        float d = deg[i];
        deg[i] = (d > 0.0f) ? (1.0f / d) : 0.0f;
    }
}

// ---------------- scatter layer-1: agg1[dst] += x[src]  (F=4) ----------------
// one thread per (edge, feature): feature dim contiguous -> coalesced L2 atomics
__global__ void k_scatter1(const int* __restrict__ src, const int* __restrict__ dst,
                           const float* __restrict__ x, float* __restrict__ agg1) {
    int t = blockIdx.x * blockDim.x + threadIdx.x;
    if (t >= NEDGES * 4) return;
    int e = t >> 2;
    int j = t & 3;
    int s = src[e];
    int d = dst[e];
    atomicAdd(&agg1[d * 4 + j], x[s * 4 + j]);
}

// ---------------- layer 1: h = relu(mean @ W1_l^T + b1 + x @ W1_r^T) --------
// One wave handles a 16-node tile via two chained V_WMMA_F32_16X16X4_F32:
//   D(16x16) = Amean(16x4) x B_l(4x16) + Ax(16x4) x B_r(4x16)
// A layout (ISA 7.12.2, 32-bit 16x4): lanes 0-15 -> K={0,1}, lanes 16-31 -> K={2,3}
// B layout symmetric; C/D: VGPR v holds rows M=v (lanes 0-15) and M=v+8 (lanes 16-31).
__global__ void k_layer1_wmma(const float* __restrict__ x,
                              const float* __restrict__ agg1,
                              const float* __restrict__ deg_inv,
                              const float* __restrict__ W1_l,  // [16,4]
                              const float* __restrict__ b1,    // [16]
                              const float* __restrict__ W1_r,  // [16,4]
                              float* __restrict__ h)           // [N,16]
{
    int tile = (blockIdx.x * blockDim.x + threadIdx.x) >> 5;
    if (tile >= NTILES) return;          // wave-uniform exit; WMMA waves stay full-EXEC
    int lane = threadIdx.x & 31;
    int row  = lane & 15;                // A: node-in-tile (M); B: out-feature (N)
    int hi   = lane >> 4;                // 0 or 1
    int k0   = hi * 2;                   // K base for this lane's A/B VGPR pair
    int node = tile * 16 + row;

    // A operands
    float di = deg_inv[node];
    v2f am, ax;
    am.x = agg1[node * 4 + k0]     * di;
    am.y = agg1[node * 4 + k0 + 1] * di;
    ax.x = x[node * 4 + k0];
    ax.y = x[node * 4 + k0 + 1];

    // B operands: B[k][n] = W[n][k]  (W row-major [16,4])
    v2f bl, br;
    bl.x = W1_l[row * 4 + k0];
    bl.y = W1_l[row * 4 + k0 + 1];
    br.x = W1_r[row * 4 + k0];
    br.y = W1_r[row * 4 + k0 + 1];

    v8f c = {};
    c = __builtin_amdgcn_wmma_f32_16x16x4_f32(false, am, false, bl, (short)0, c, false, false);
    c = __builtin_amdgcn_wmma_f32_16x16x4_f32(false, ax, false, br, (short)0, c, false, false);

    float bias = b1[row];                // column (out feature) = row index for this lane
    #pragma unroll
    for (int v = 0; v < 8; ++v) {
        float val = c[v] + bias;
        val = fmaxf(val, 0.0f);          // ReLU
        int m = v + hi * 8;              // node row within tile
        h[(tile * 16 + m) * 16 + row] = val;
    }
}

// ---------------- scatter layer-2: agg2[dst] += h[src]  (F=16) ----------------
__global__ void k_scatter2(const int* __restrict__ src, const int* __restrict__ dst,
                           const float* __restrict__ h, float* __restrict__ agg2) {
    int t = blockIdx.x * blockDim.x + threadIdx.x;
    if (t >= NEDGES * 16) return;
    int e = t >> 4;
    int j = t & 15;
    int s = src[e];
    int d = dst[e];
    atomicAdd(&agg2[d * 16 + j], h[s * 16 + j]);
}

// ---------------- layer 2: out = mean2 @ W2_l^T + b2 + h @ W2_r^T (16->1) ----
__global__ void k_layer2(const float* __restrict__ h,
                         const float* __restrict__ agg2,
                         const float* __restrict__ deg_inv,
                         const float* __restrict__ W2_l,  // [16]
                         const float* __restrict__ b2,    // [1]
                         const float* __restrict__ W2_r,  // [16]
                         float* __restrict__ out) {
    int i = blockIdx.x * blockDim.x + threadIdx.x;
    if (i >= NNODES) return;
    float di  = deg_inv[i];
    float acc = b2[0];
    const float4* a4 = (const float4*)(agg2 + i * 16);
    const float4* h4 = (const float4*)(h + i * 16);
    #pragma unroll
    for (int q = 0; q < 4; ++q) {
        float4 av = a4[q];
        float4 hv = h4[q];
        acc += (av.x * di) * W2_l[q * 4 + 0] + hv.x * W2_r[q * 4 + 0];
        acc += (av.y * di) * W2_l[q * 4 + 1] + hv.y * W2_r[q * 4 + 1];
        acc += (av.z * di) * W2_l[q * 4 + 2] + hv.z * W2_r[q * 4 + 2];
        acc += (av.w * di) * W2_l[q * 4 + 3] + hv.w * W2_r[q * 4 + 3];
    }
    out[i] = acc;
}

extern "C" void kernel_launch(void* const* d_in, const int* in_sizes, int n_in,
                              void* d_out, int out_size, void* d_ws, size_t ws_size,
                              hipStream_t stream) {
    const float* x    = (const float*)d_in[0];          // [N,4]
    const int*   ei   = (const int*)d_in[1];            // [2,E]
    const float* W1_l = (const float*)d_in[2];          // [16,4]
    const float* b1   = (const float*)d_in[3];          // [16]
    const float* W1_r = (const float*)d_in[4];          // [16,4]
    const float* W2_l = (const float*)d_in[5];          // [1,16]
    const float* b2   = (const float*)d_in[6];          // [1]
    const float* W2_r = (const float*)d_in[7];          // [1,16]
    float* out = (float*)d_out;

    const int* src = ei;
    const int* dst = ei + NEDGES;

    // workspace layout (floats): deg[N] | agg1[4N] | agg2[16N] | h[16N]
    float* ws   = (float*)d_ws;
    float* deg  = ws;                       // becomes deg_inv in place
    float* agg1 = ws + (size_t)NNODES;
    float* agg2 = ws + (size_t)NNODES * 5;
    float* h    = ws + (size_t)NNODES * 21;

    const int B = 256;

    // 1) zero deg + agg1 + agg2  (21N contiguous floats; h fully overwritten later)
    {
        int n = NNODES * 21;
        k_zero<<<(n + B - 1) / B, B, 0, stream>>>(ws, n);
    }
    // 2) degree
    k_degree<<<(NEDGES + B - 1) / B, B, 0, stream>>>(dst, deg);
    // 3) deg -> 1/deg
    k_deg_inv<<<(NNODES + B - 1) / B, B, 0, stream>>>(deg);
    // 4) neighbor sum, layer 1
    {
        int n = NEDGES * 4;
        k_scatter1<<<(n + B - 1) / B, B, 0, stream>>>(src, dst, x, agg1);
    }
    // 5) layer 1 dense transform via WMMA (8 tiles / 256-thread block)
    {
        int blocks = (NTILES + 7) / 8;
        k_layer1_wmma<<<blocks, B, 0, stream>>>(x, agg1, deg, W1_l, b1, W1_r, h);
    }
    // 6) neighbor sum, layer 2
    {
        int n = NEDGES * 16;
        k_scatter2<<<(n + B - 1) / B, B, 0, stream>>>(src, dst, h, agg2);
    }
    // 7) layer 2 (16->1 dot products, VALU)
    k_layer2<<<(NNODES + B - 1) / B, B, 0, stream>>>(h, agg2, deg, W2_l, b2, W2_r, out);
}